// AffineAugmentWithB0_2087354106269
// MI455X (gfx1250) — compile-verified
//
#include <hip/hip_runtime.h>

// Problem constants (match reference): B=8, D=H=W=160, C=1
#define Bv 8
#define Dv 160
#define Hv 160
#define Wv 160
static constexpr size_t SLICE = (size_t)Hv * Wv;        // one z-plane
static constexpr size_t VOL   = (size_t)Dv * Hv * Wv;   // one batch volume

// Trilinear sample of a (Dv,Hv,Wv) f32 volume at (lz,ly,lx) in voxel coords.
// Matches the reference: valid iff every coord in [0, dim-1], else fill 0;
// coords clipped, i_lo = floor, i_hi = i_lo+1 (edge handled by clamping i_lo
// to dim-2 and recomputing the fractional weight, which is mathematically
// identical to the reference's min(i_lo+1, dim-1) with zero hi-weight).
__device__ __forceinline__ float trilinear160(const float* __restrict__ vol,
                                              float lz, float ly, float lx) {
    const float mz = (float)(Dv - 1), my = (float)(Hv - 1), mx = (float)(Wv - 1);
    const bool valid = (lz >= 0.0f) & (lz <= mz) &
                       (ly >= 0.0f) & (ly <= my) &
                       (lx >= 0.0f) & (lx <= mx);

    lz = fminf(fmaxf(lz, 0.0f), mz);
    ly = fminf(fmaxf(ly, 0.0f), my);
    lx = fminf(fmaxf(lx, 0.0f), mx);

    int iz = min((int)floorf(lz), Dv - 2);
    int iy = min((int)floorf(ly), Hv - 2);
    int ix = min((int)floorf(lx), Wv - 2);

    const float wz1 = lz - (float)iz, wz0 = 1.0f - wz1;
    const float wy1 = ly - (float)iy, wy0 = 1.0f - wy1;
    const float wx1 = lx - (float)ix, wx0 = 1.0f - wx1;

    const float* p = vol + ((size_t)iz * Hv + (size_t)iy) * Wv + (size_t)ix;
    const float* q = p + SLICE;  // iz+1 plane (in-bounds: iz <= Dv-2)

    // 8 gathers; adjacent lanes gather adjacent addresses -> coalesced.
    const float v000 = p[0],      v001 = p[1];
    const float v010 = p[Wv],     v011 = p[Wv + 1];
    const float v100 = q[0],      v101 = q[1];
    const float v110 = q[Wv],     v111 = q[Wv + 1];

    const float r =
        wz0 * (wy0 * (wx0 * v000 + wx1 * v001) +
               wy1 * (wx0 * v010 + wx1 * v011)) +
        wz1 * (wy0 * (wx0 * v100 + wx1 * v101) +
               wy1 * (wx0 * v110 + wx1 * v111));

    return valid ? r : 0.0f;
}

// Pass 1: affine resample. One block = one W-row (160 threads = 5 wave32s).
// grid = (Hv, Dv, Bv). Output stays RT so it remains resident in the 192MB L2
// for pass 2's gather.
__global__ __launch_bounds__(Wv)
void affine_resample_kernel(const float* __restrict__ im,
                            const float* __restrict__ affine,
                            float* __restrict__ out) {
    const int x = threadIdx.x;   // W
    const int y = blockIdx.x;    // H
    const int z = blockIdx.y;    // D
    const int b = blockIdx.z;    // B

    // 3x4 row-major affine for this batch; uniform -> scalar loads.
    const float* A = affine + b * 12;
    const float fz = (float)z, fy = (float)y, fx = (float)x;

    // loc_i = A[i][0]*z + A[i][1]*y + A[i][2]*x + A[i][3]
    const float lz = fmaf(A[0], fz, fmaf(A[1], fy, fmaf(A[2],  fx, A[3])));
    const float ly = fmaf(A[4], fz, fmaf(A[5], fy, fmaf(A[6],  fx, A[7])));
    const float lx = fmaf(A[8], fz, fmaf(A[9], fy, fmaf(A[10], fx, A[11])));

    const float v = trilinear160(im + (size_t)b * VOL, lz, ly, lx);

    out[(size_t)b * VOL + (size_t)z * SLICE + (size_t)y * Wv + (size_t)x] = v;
}

// Pass 2: B0 distortion warp of the pass-1 result.
// field is a pure stream (read once) -> nontemporal load + gfx1250
// global_prefetch of the next z-plane; distorted is write-once ->
// nontemporal store. Both keep the L2 free for the `transformed` gather.
__global__ __launch_bounds__(Wv)
void b0_distort_kernel(const float* __restrict__ transformed,
                       const float* __restrict__ field,
                       const float* __restrict__ b0_dir,
                       float* __restrict__ out) {
    const int x = threadIdx.x;
    const int y = blockIdx.x;
    const int z = blockIdx.y;
    const int b = blockIdx.z;

    const size_t idx = (size_t)b * VOL + (size_t)z * SLICE + (size_t)y * Wv + (size_t)x;

    const float f = __builtin_nontemporal_load(&field[idx]);
    if (z + 1 < Dv) {
        // warm L2 for the next z-plane of the streamed field (global_prefetch_b8)
        __builtin_prefetch(&field[idx + SLICE], 0, 1);
    }

    const float dz = b0_dir[b * 3 + 0];
    const float dy = b0_dir[b * 3 + 1];
    const float dx = b0_dir[b * 3 + 2];

    const float lz = fmaf(f, dz, (float)z);
    const float ly = fmaf(f, dy, (float)y);
    const float lx = fmaf(f, dx, (float)x);

    const float v = trilinear160(transformed + (size_t)b * VOL, lz, ly, lx);

    __builtin_nontemporal_store(v, &out[idx]);
}

extern "C" void kernel_launch(void* const* d_in, const int* in_sizes, int n_in,
                              void* d_out, int out_size, void* d_ws, size_t ws_size,
                              hipStream_t stream) {
    (void)in_sizes; (void)n_in; (void)out_size; (void)d_ws; (void)ws_size;

    const float* im     = (const float*)d_in[0];  // (B,D,H,W,1) f32
    const float* affine = (const float*)d_in[1];  // (B,3,4)     f32
    const float* field  = (const float*)d_in[2];  // (B,D,H,W)   f32
    const float* b0dir  = (const float*)d_in[3];  // (B,3)       f32

    float* transformed = (float*)d_out;                 // first output of tuple
    float* distorted   = transformed + (size_t)Bv * VOL; // second output of tuple

    dim3 block(Wv, 1, 1);      // 160 threads = 5 wave32s, one W-row per block
    dim3 grid(Hv, Dv, Bv);     // (160, 160, 8)

    affine_resample_kernel<<<grid, block, 0, stream>>>(im, affine, transformed);
    b0_distort_kernel<<<grid, block, 0, stream>>>(transformed, field, b0dir, distorted);
}